// DSVF_67851893342354
// MI455X (gfx1250) — compile-verified
//
#include <hip/hip_runtime.h>
#include <math.h>

typedef __attribute__((ext_vector_type(2))) float v2f;
typedef __attribute__((ext_vector_type(8))) float v8f;

#define BB    512      // rows
#define TT    32768    // samples per row
#define CC    128      // chunks per row
#define LL    256      // samples per chunk
#define BLKT  16       // WMMA block length in time
#define NBLK  (LL / BLKT)
#define TILES (BB / 16)

// coef layout (floats):
//  [0]=b0 [1]=a1 [2]=a2 [3]=v0 [4]=v1
//  [8..11]  = M16 (m00 m01 m10 m11)
//  [12..15] = ML = M^LL
//  [16..31] = h[0..15]   (impulse response)
//  [32..47] = G0[t]      (first row of M^t, col 0)
//  [48..63] = G1[t]      (first row of M^t, col 1)
//  [64..79] = K0[tau]    ((M^{15-tau} v).x)
//  [80..95] = K1[tau]    ((M^{15-tau} v).y)

__global__ void dsvf_setup(const float* gp, const float* rp, const float* mhpp,
                           const float* mbpp, const float* mlpp, float* coef) {
  if (threadIdx.x != 0 || blockIdx.x != 0) return;
  float graw = gp[0], rraw = rp[0];
  float m_hp = mhpp[0], m_bp = mbpp[0], m_lp = mlpp[0];
  float sig = 1.0f / (1.0f + expf(-graw));
  float gg  = tanf(3.14159265358979323846f * sig * 0.5f);
  float rr  = (rraw > 20.0f) ? rraw : log1pf(expf(rraw));
  float g2  = gg * gg;
  float b0u = g2 * m_lp + gg * m_bp + m_hp;
  float b1u = 2.0f * g2 * m_lp - 2.0f * m_hp;
  float b2u = g2 * m_lp - gg * m_bp + m_hp;
  float a0  = g2 + 2.0f * rr * gg + 1.0f;
  float a1u = 2.0f * g2 - 2.0f;
  float a2u = g2 - 2.0f * rr * gg + 1.0f;
  float inv = 1.0f / a0;
  float b0 = b0u * inv, b1 = b1u * inv, b2 = b2u * inv;
  float a1 = a1u * inv, a2 = a2u * inv;
  // state form: s_t = M s_{t-1} + v x_t ;  y_t = b0 x_t + z1_{t-1}
  float v0 = b1 - a1 * b0, v1 = b2 - a2 * b0;
  float m00 = -a1, m01 = 1.0f, m10 = -a2, m11 = 0.0f;
  coef[0] = b0; coef[1] = a1; coef[2] = a2; coef[3] = v0; coef[4] = v1;
  // impulse response: h0 = b0, h_d = (M^{d-1} v).x
  coef[16] = b0;
  float w0 = v0, w1 = v1;
  for (int d = 1; d < 16; ++d) {
    coef[16 + d] = w0;
    float n0 = m00 * w0 + m01 * w1, n1 = m10 * w0 + m11 * w1;
    w0 = n0; w1 = n1;
  }
  // G[t] = first row of M^t (t=0..15); ends with P = M^16
  float p00 = 1.f, p01 = 0.f, p10 = 0.f, p11 = 1.f;
  for (int t = 0; t < 16; ++t) {
    coef[32 + t] = p00; coef[48 + t] = p01;
    float n00 = m00 * p00 + m01 * p10, n01 = m00 * p01 + m01 * p11;
    float n10 = m10 * p00 + m11 * p10, n11 = m10 * p01 + m11 * p11;
    p00 = n00; p01 = n01; p10 = n10; p11 = n11;
  }
  coef[8] = p00; coef[9] = p01; coef[10] = p10; coef[11] = p11;   // M16
  // K[tau] = M^{15-tau} v
  w0 = v0; w1 = v1;
  for (int tau = 15; tau >= 0; --tau) {
    coef[64 + tau] = w0; coef[80 + tau] = w1;
    float n0 = m00 * w0 + m01 * w1, n1 = m10 * w0 + m11 * w1;
    w0 = n0; w1 = n1;
  }
  // ML = M^LL = (M16)^16 via 4 squarings
  float q00 = p00, q01 = p01, q10 = p10, q11 = p11;
  for (int s = 0; s < 4; ++s) {
    float n00 = q00 * q00 + q01 * q10, n01 = q00 * q01 + q01 * q11;
    float n10 = q10 * q00 + q11 * q10, n11 = q10 * q01 + q11 * q11;
    q00 = n00; q01 = n01; q10 = n10; q11 = n11;
  }
  coef[12] = q00; coef[13] = q01; coef[14] = q10; coef[15] = q11;
}

// Pass 1: zero-state chunk-final states. One thread per (row, chunk).
__global__ void __launch_bounds__(256)
dsvf_pass1(const float* __restrict__ x, const float* __restrict__ coef,
           float* __restrict__ fstate) {
  int tid = blockIdx.x * blockDim.x + threadIdx.x;   // 0 .. BB*CC-1
  int row = tid >> 7;          // / CC
  int c   = tid & (CC - 1);
  float a1 = coef[1], a2 = coef[2], v0 = coef[3], v1 = coef[4];
  const float4* p = (const float4*)(x + (size_t)row * TT + (size_t)c * LL);
  float z1 = 0.f, z2 = 0.f;
#pragma unroll 4
  for (int i = 0; i < LL / 4; ++i) {
    float4 xv = p[i];
    float xs[4] = {xv.x, xv.y, xv.z, xv.w};
#pragma unroll
    for (int j = 0; j < 4; ++j) {
      float t   = fmaf(v0, xs[j], z2);
      float nz1 = fmaf(-a1, z1, t);
      float nz2 = fmaf(v1, xs[j], -a2 * z1);
      z1 = nz1; z2 = nz2;
    }
  }
  fstate[2 * tid]     = z1;
  fstate[2 * tid + 1] = z2;
}

// Pass 2: per-row scan over chunks: s_{c+1} = ML * s_c + f_c ; record s_c.
__global__ void dsvf_pass2(const float* __restrict__ coef,
                           const float* __restrict__ fstate,
                           float* __restrict__ sstate) {
  int row = blockIdx.x * blockDim.x + threadIdx.x;
  if (row >= BB) return;
  float l00 = coef[12], l01 = coef[13], l10 = coef[14], l11 = coef[15];
  float s1 = 0.f, s2 = 0.f;
  for (int c = 0; c < CC; ++c) {
    int idx = (row * CC + c) * 2;
    sstate[idx] = s1; sstate[idx + 1] = s2;
    float f1 = fstate[idx], f2 = fstate[idx + 1];
    float n1 = l00 * s1 + l01 * s2 + f1;
    float n2 = l10 * s1 + l11 * s2 + f2;
    s1 = n1; s2 = n2;
  }
}

// Pass 3: one wave per (16-row tile, chunk). Per 16-sample block:
//   Y(16x16) = X(16x16) * H^T + S0(16x2,padded) * G^T   via 5 x v_wmma_f32_16x16x4_f32
//   s_new = M16 * s + K . x_block  (cross-lane-half reduce via shfl_xor 16)
__global__ void __launch_bounds__(256)
dsvf_pass3(const float* __restrict__ x, const float* __restrict__ coef,
           const float* __restrict__ sstate, float* __restrict__ y) {
  int lane = threadIdx.x & 31;
  int wave = blockIdx.x * (blockDim.x >> 5) + (threadIdx.x >> 5);
  int tile = wave / CC;
  int c    = wave % CC;
  int r0   = tile * 16;
  int half = lane >> 4;     // 0: K even pair, 1: K odd pair (A/B lane-half split)
  int tn   = lane & 15;     // N (time within block) for B/C/D; row for A

  // B matrices for the Toeplitz product: B_q[k][t] = h[t - (4q+k)] (t >= 4q+k)
  v2f Bq[4];
#pragma unroll
  for (int q = 0; q < 4; ++q) {
    int ta = 4 * q + half * 2;
    float e0 = (tn >= ta)     ? coef[16 + tn - ta]     : 0.f;
    float e1 = (tn >= ta + 1) ? coef[16 + tn - ta - 1] : 0.f;
    Bq[q] = (v2f){e0, e1};
  }
  // State-term B: rows 0,1 = G^T, rows 2,3 = 0
  v2f Bg = (v2f){half ? 0.f : coef[32 + tn], half ? 0.f : coef[48 + tn]};

  // Per-lane K coefficients at this lane's tau positions
  float k0[8], k1[8];
#pragma unroll
  for (int q = 0; q < 4; ++q)
#pragma unroll
    for (int j = 0; j < 2; ++j) {
      int tau = 4 * q + half * 2 + j;
      k0[2 * q + j] = coef[64 + tau];
      k1[2 * q + j] = coef[80 + tau];
    }
  float m00 = coef[8], m01 = coef[9], m10 = coef[10], m11 = coef[11];

  int rowA = r0 + tn;                 // this lane's A-matrix row
  float z1 = 0.f, z2 = 0.f;
  if (half == 0) {
    int idx = (rowA * CC + c) * 2;
    z1 = sstate[idx]; z2 = sstate[idx + 1];
  }

  const float* xrow = x + (size_t)rowA * TT;
  int tbase = c * LL;
  for (int blk = 0; blk < NBLK; ++blk) {
    int tb = tbase + blk * BLKT;
    v2f A[4];
#pragma unroll
    for (int q = 0; q < 4; ++q) {
      int t = tb + 4 * q + half * 2;
      A[q] = *(const v2f*)(xrow + t);   // 8-byte load: this lane's (K, K+1) pair
    }
    v2f As = (v2f){half ? 0.f : z1, half ? 0.f : z2};  // S0 padded to 16x4

    v8f acc = {};
    acc = __builtin_amdgcn_wmma_f32_16x16x4_f32(false, As,   false, Bg,    (short)0, acc, false, false);
    acc = __builtin_amdgcn_wmma_f32_16x16x4_f32(false, A[0], false, Bq[0], (short)0, acc, false, false);
    acc = __builtin_amdgcn_wmma_f32_16x16x4_f32(false, A[1], false, Bq[1], (short)0, acc, false, false);
    acc = __builtin_amdgcn_wmma_f32_16x16x4_f32(false, A[2], false, Bq[2], (short)0, acc, false, false);
    acc = __builtin_amdgcn_wmma_f32_16x16x4_f32(false, A[3], false, Bq[3], (short)0, acc, false, false);

    // Store D: VGPR g holds Y[r0+g+half*8][tb+tn]
#pragma unroll
    for (int gdx = 0; gdx < 8; ++gdx) {
      int rr = r0 + gdx + half * 8;
      y[(size_t)rr * TT + tb + tn] = acc[gdx];
    }

    // State update: s_new = M16 * s + K . x_block (reduce across lane halves)
    float p0 = 0.f, p1 = 0.f;
#pragma unroll
    for (int q = 0; q < 4; ++q) {
      p0 = fmaf(k0[2 * q + 0], A[q].x, p0);
      p0 = fmaf(k0[2 * q + 1], A[q].y, p0);
      p1 = fmaf(k1[2 * q + 0], A[q].x, p1);
      p1 = fmaf(k1[2 * q + 1], A[q].y, p1);
    }
    p0 += __shfl_xor(p0, 16, 32);
    p1 += __shfl_xor(p1, 16, 32);
    float n1 = m00 * z1 + m01 * z2 + p0;
    float n2 = m10 * z1 + m11 * z2 + p1;
    z1 = n1; z2 = n2;
  }
}

extern "C" void kernel_launch(void* const* d_in, const int* in_sizes, int n_in,
                              void* d_out, int out_size, void* d_ws, size_t ws_size,
                              hipStream_t stream) {
  const float* x   = (const float*)d_in[0];
  const float* g   = (const float*)d_in[1];
  const float* r   = (const float*)d_in[2];
  const float* mhp = (const float*)d_in[3];
  const float* mbp = (const float*)d_in[4];
  const float* mlp = (const float*)d_in[5];
  float* y = (float*)d_out;

  char*  ws     = (char*)d_ws;
  float* coef   = (float*)ws;                                        // 384 B used, 1 KB reserved
  float* fstate = (float*)(ws + 1024);                               // 512 KB
  float* sstate = (float*)(ws + 1024 + (size_t)BB * CC * 2 * 4);     // 512 KB

  dsvf_setup<<<1, 32, 0, stream>>>(g, r, mhp, mbp, mlp, coef);
  dsvf_pass1<<<(BB * CC) / 256, 256, 0, stream>>>(x, coef, fstate);
  dsvf_pass2<<<2, 256, 0, stream>>>(coef, fstate, sstate);
  dsvf_pass3<<<(TILES * CC) / 8, 256, 0, stream>>>(x, coef, sstate, y);
}